// GraphKernelNetwork_10874857193550
// MI455X (gfx1250) — compile-verified
//
#include <hip/hip_runtime.h>
#include <math.h>

#define GAMMA 0.7f
#define NF 8

typedef __attribute__((ext_vector_type(2))) float v2f;
typedef __attribute__((ext_vector_type(8))) float v8f;

__device__ __forceinline__ float softplusf(float x) {
  return (x > 20.f) ? x : log1pf(expf(x));
}
__device__ __forceinline__ float gelu_exact(float x) {
  // gelu approximate=False: x * 0.5 * (1 + erf(x / sqrt(2)))
  return 0.5f * x * (1.f + erff(x * 0.70710678118654752440f));
}

// ---------------------------------------------------------------- zero fill
__global__ void zero_f32(float* p, int nelems) {
  for (int i = blockIdx.x * blockDim.x + threadIdx.x; i < nelems;
       i += gridDim.x * blockDim.x)
    p[i] = 0.f;
}

// ------------------------------------------------ weight pad + transpose
// W: K x N row-major  ->  Wt: N x Kpad row-major, zero padded rows K..Kpad-1.
// Makes the per-lane B-operand fetch in the WMMA GEMM a contiguous b64 load.
__global__ void pad_transpose(const float* __restrict__ W, int K, int N, int Kpad,
                              float* __restrict__ Wt) {
  int i = blockIdx.x * blockDim.x + threadIdx.x;
  int total = N * Kpad;
  if (i >= total) return;
  int nIdx = i / Kpad, k = i - nIdx * Kpad;
  Wt[i] = (k < K) ? W[(size_t)k * N + nIdx] : 0.f;
}

// ------------------------------------------------- per-filter trace precompute
// For each adjacency-filter tensor W (F,k,k):
//   S  = softplus(W); Ssym = 0.5*(S+S^T)*(1-I); Sf = rownorm(Ssym)
//   store tr(Sf), tr(Sf^2), tr(Sf^3)   (Kronecker identity: tr((T(x)S)^p)=tr(T^p)tr(S^p))
// traces layout: [tensor(4)][filter(8)][p(3)]; tensor: 0=tri0 1=cyc0 2=tri1 3=cyc1
__global__ void filter_prep(const float* __restrict__ Wt0, const float* __restrict__ Wc0,
                            const float* __restrict__ g0,
                            const float* __restrict__ Wt1, const float* __restrict__ Wc1,
                            const float* __restrict__ g1,
                            float* __restrict__ traces, float* __restrict__ gp) {
  int t = threadIdx.x;  // 0..31
  if (t < 32) {
    int tensor = t >> 3, f = t & 7;
    const float* W = (tensor == 0) ? Wt0 : (tensor == 1) ? Wc0 : (tensor == 2) ? Wt1 : Wc1;
    const int k = (tensor & 1) ? 4 : 3;
    float S[4][4], Sf[4][4];
    for (int a = 0; a < k; ++a)
      for (int b = 0; b < k; ++b)
        S[a][b] = softplusf(W[f * k * k + a * k + b]);
    for (int a = 0; a < k; ++a) {
      float rs = 0.f;
      for (int b = 0; b < k; ++b) {
        float v = (a == b) ? 0.f : 0.5f * (S[a][b] + S[b][a]);
        Sf[a][b] = v;
        rs += v;
      }
      float inv = 1.f / fmaxf(rs, 1e-12f);
      for (int b = 0; b < k; ++b) Sf[a][b] *= inv;
    }
    float s1 = 0.f;
    for (int a = 0; a < k; ++a) s1 += Sf[a][a];
    float S2[4][4];
    for (int a = 0; a < k; ++a)
      for (int b = 0; b < k; ++b) {
        float acc = 0.f;
        for (int c = 0; c < k; ++c) acc += Sf[a][c] * Sf[c][b];
        S2[a][b] = acc;
      }
    float s2 = 0.f;
    for (int a = 0; a < k; ++a) s2 += S2[a][a];
    float s3 = 0.f;
    for (int a = 0; a < k; ++a)
      for (int b = 0; b < k; ++b) s3 += S2[a][b] * Sf[b][a];
    traces[t * 3 + 0] = s1;
    traces[t * 3 + 1] = s2;
    traces[t * 3 + 2] = s3;
  }
  if (t == 0) { for (int p = 0; p < 3; ++p) gp[p] = softplusf(g0[p]); }
  if (t == 1) { for (int p = 0; p < 3; ++p) gp[3 + p] = softplusf(g1[p]); }
}

// ---------------------------------------------------------------- row sums of A
__global__ void row_sum(const float* __restrict__ A, int n,
                        float* __restrict__ phi0, int ld0, int c0,
                        float* __restrict__ phi1, int ld1, int c1,
                        float* __restrict__ out, int ldo, int co) {
  int row = blockIdx.x;
  const float4* Ar = (const float4*)(A + (size_t)row * n);
  int n4 = n >> 2;
  float s = 0.f;
  for (int i = threadIdx.x; i < n4; i += blockDim.x) {
    float4 v = Ar[i];
    s += v.x + v.y + v.z + v.w;
  }
  __shared__ float sm[256];
  sm[threadIdx.x] = s;
  __syncthreads();
  for (int off = blockDim.x >> 1; off > 0; off >>= 1) {
    if (threadIdx.x < off) sm[threadIdx.x] += sm[threadIdx.x + off];
    __syncthreads();
  }
  if (threadIdx.x == 0) {
    float v = sm[0];
    phi0[(size_t)row * ld0 + c0] = v;
    phi1[(size_t)row * ld1 + c1] = v;
    out[(size_t)row * ldo + co] = v;
  }
}

// ------------------------------------------------------------- motif channel
// Kronecker factorization: sims[i,f] = sum_p gamma^p*gp[p-1]*tr(Tm^p)*tr(Sf_f^p)
// Tm depends only on (A, idx) -> shared across both layers.
template <int K>
__global__ void motif_kernel(const float* __restrict__ A, int n,
                             const int* __restrict__ idx, int I,
                             const float* __restrict__ tr0, const float* __restrict__ gp0,
                             const float* __restrict__ tr1, const float* __restrict__ gp1,
                             float* __restrict__ phi0, int ld0, int c0,
                             float* __restrict__ phi1, int ld1, int c1) {
  int i = blockIdx.x * blockDim.x + threadIdx.x;
  if (i >= I) return;
  int id[K];
#pragma unroll
  for (int j = 0; j < K; ++j) id[j] = idx[i * K + j];
  float T[K][K];
#pragma unroll
  for (int a = 0; a < K; ++a) {
    const float* Ar = A + (size_t)id[a] * n;
    float s = 0.f;
#pragma unroll
    for (int c = 0; c < K; ++c) {
      float v = Ar[id[c]];
      T[a][c] = v;
      s += v;
    }
    float inv = 1.f / fmaxf(s, 1e-12f);
#pragma unroll
    for (int c = 0; c < K; ++c) T[a][c] *= inv;
  }
  float t1 = 0.f;
#pragma unroll
  for (int a = 0; a < K; ++a) t1 += T[a][a];
  float M2[K][K];
#pragma unroll
  for (int a = 0; a < K; ++a)
#pragma unroll
    for (int b = 0; b < K; ++b) {
      float acc = 0.f;
#pragma unroll
      for (int c = 0; c < K; ++c) acc += T[a][c] * T[c][b];
      M2[a][b] = acc;
    }
  float t2 = 0.f;
#pragma unroll
  for (int a = 0; a < K; ++a) t2 += M2[a][a];
  float t3 = 0.f;
#pragma unroll
  for (int a = 0; a < K; ++a)
#pragma unroll
    for (int b = 0; b < K; ++b) t3 += M2[a][b] * T[b][a];

#pragma unroll
  for (int layer = 0; layer < 2; ++layer) {
    const float* tr = layer ? tr1 : tr0;
    const float* gv = layer ? gp1 : gp0;
    float w1 = GAMMA * gv[0] * t1;
    float w2 = GAMMA * GAMMA * gv[1] * t2;
    float w3 = GAMMA * GAMMA * GAMMA * gv[2] * t3;
    float sims[NF], mx = -1e30f;
#pragma unroll
    for (int f = 0; f < NF; ++f) {
      sims[f] = w1 * tr[f * 3 + 0] + w2 * tr[f * 3 + 1] + w3 * tr[f * 3 + 2];
      mx = fmaxf(mx, sims[f]);
    }
    float e[NF], den = 0.f;
#pragma unroll
    for (int f = 0; f < NF; ++f) {  // TAU == 1.0
      e[f] = expf(sims[f] - mx);
      den += e[f];
    }
    float invden = 1.f / den;
    float contrib[NF];
#pragma unroll
    for (int f = 0; f < NF; ++f) contrib[f] = e[f] * invden * sims[f] * (1.f / (float)K);
    float* phi = layer ? phi1 : phi0;
    int ld = layer ? ld1 : ld0;
    int co = layer ? c1 : c0;
#pragma unroll
    for (int j = 0; j < K; ++j) {
      float* dst = phi + (size_t)id[j] * ld + co;
#pragma unroll
      for (int f = 0; f < NF; ++f) atomicAdd(dst + f, contrib[f]);
    }
  }
}

// ------------------------------------------------------------- layernorm (rowwise)
__global__ void layernorm_rows(float* __restrict__ X, int ld, int nf,
                               const float* __restrict__ lng, const float* __restrict__ lnb) {
  int row = blockIdx.x;
  float* xr = X + (size_t)row * ld;
  float s = 0.f, s2 = 0.f;
  for (int c = threadIdx.x; c < nf; c += 32) {
    float v = xr[c];
    s += v;
    s2 += v * v;
  }
  for (int o = 16; o > 0; o >>= 1) {
    s += __shfl_xor(s, o, 32);
    s2 += __shfl_xor(s2, o, 32);
  }
  float mean = s / (float)nf;
  float var = s2 / (float)nf - mean * mean;
  float inv = rsqrtf(var + 1e-5f);
  for (int c = threadIdx.x; c < nf; c += 32) xr[c] = (xr[c] - mean) * inv * lng[c] + lnb[c];
}

// ----------------------------------------- WMMA f32 GEMM: Y = gelu(X @ Wt^T + b)
// One wave per 16x16 C tile; K chained in steps of 4 via V_WMMA_F32_16X16X4_F32.
// A layout (16x4): lanes 0-15 hold K={k,k+1}, lanes 16-31 hold K={k+2,k+3}, M=lane&15.
// B layout (4x16): mirrored by column. C: VGPR r -> M = r + 8*(lane>=16), N = lane&15.
// Wt is pre-transposed & zero-padded (N x Kpad) so both operand fetches are b64.
__global__ void gemm_wmma_gelu(const float* __restrict__ X, int ldx,
                               const float* __restrict__ Wt, int Kpad,
                               const float* __restrict__ bias,
                               float* __restrict__ Y, int ldy) {
  int lane = threadIdx.x;
  int half = lane >> 4, lr = lane & 15;
  int tileM = blockIdx.x * 16, tileN = blockIdx.y * 16;
  int col = tileN + lr;
  const float* xrow = X + (size_t)(tileM + lr) * ldx + 2 * half;
  const float* brow = Wt + (size_t)col * Kpad + 2 * half;
  v8f c = {};
#pragma unroll 4
  for (int k = 0; k < Kpad; k += 4) {
    v2f a, b;
    a.x = xrow[k];
    a.y = xrow[k + 1];
    b.x = brow[k];
    b.y = brow[k + 1];
    c = __builtin_amdgcn_wmma_f32_16x16x4_f32(false, a, false, b, (short)0, c, false, false);
  }
  float bv = bias[col];
#pragma unroll
  for (int r = 0; r < 8; ++r) {
    int rowo = tileM + r + 8 * half;
    Y[(size_t)rowo * ldy + col] = gelu_exact(c[r] + bv);
  }
}

// ------------------------------- WMMA f32 GEMM: Y = X @ Wt^T + b (dual-destination)
__global__ void gemm_wmma_bias(const float* __restrict__ X, int ldx,
                               const float* __restrict__ Wt, int Kpad,
                               const float* __restrict__ bias,
                               float* __restrict__ Y0, int ldy0, int cof0,
                               float* __restrict__ Y1, int ldy1, int cof1) {
  int lane = threadIdx.x;
  int half = lane >> 4, lr = lane & 15;
  int tileM = blockIdx.x * 16, tileN = blockIdx.y * 16;
  int col = tileN + lr;
  const float* xrow = X + (size_t)(tileM + lr) * ldx + 2 * half;
  const float* brow = Wt + (size_t)col * Kpad + 2 * half;
  v8f c = {};
#pragma unroll 4
  for (int k = 0; k < Kpad; k += 4) {
    v2f a, b;
    a.x = xrow[k];
    a.y = xrow[k + 1];
    b.x = brow[k];
    b.y = brow[k + 1];
    c = __builtin_amdgcn_wmma_f32_16x16x4_f32(false, a, false, b, (short)0, c, false, false);
  }
  float bv = bias[col];
#pragma unroll
  for (int r = 0; r < 8; ++r) {
    int rowo = tileM + r + 8 * half;
    float v = c[r] + bv;
    Y0[(size_t)rowo * ldy0 + cof0 + col] = v;
    if (Y1) Y1[(size_t)rowo * ldy1 + cof1 + col] = v;
  }
}

extern "C" void kernel_launch(void* const* d_in, const int* in_sizes, int n_in,
                              void* d_out, int out_size, void* d_ws, size_t ws_size,
                              hipStream_t stream) {
  const float* A    = (const float*)d_in[0];
  const int*   tri  = (const int*)d_in[1];
  const int*   cyc  = (const int*)d_in[2];
  const float* Wt0  = (const float*)d_in[3];
  const float* Wc0  = (const float*)d_in[4];
  const float* g0   = (const float*)d_in[5];
  const float* lng0 = (const float*)d_in[6];
  const float* lnb0 = (const float*)d_in[7];
  const float* w10  = (const float*)d_in[8];
  const float* b10  = (const float*)d_in[9];
  const float* w20  = (const float*)d_in[10];
  const float* b20  = (const float*)d_in[11];
  const float* Wt1  = (const float*)d_in[12];
  const float* Wc1  = (const float*)d_in[13];
  const float* g1   = (const float*)d_in[14];
  const float* lng1 = (const float*)d_in[15];
  const float* lnb1 = (const float*)d_in[16];
  const float* w11  = (const float*)d_in[17];
  const float* b11  = (const float*)d_in[18];
  const float* w21  = (const float*)d_in[19];
  const float* b21  = (const float*)d_in[20];
  float* out = (float*)d_out;

  const int n = (int)lround(sqrt((double)in_sizes[0]));  // 4096
  const int HID = 128, D = 128;
  const int NF0 = 17, NF1 = 145;     // actual LayerNorm feature widths
  const int LD0 = 20, LD1 = 148;     // K padded to multiple of 4
  const int OLD = 2 * D + 1;         // output row stride = 257
  const int I_tri = in_sizes[1] / 3;
  const int I_cyc = in_sizes[2] / 4;

  // workspace layout (floats)
  float* Phi0   = (float*)d_ws;                  // n * LD0
  float* Phi1   = Phi0 + (size_t)n * LD0;        // n * LD1
  float* Hid    = Phi1 + (size_t)n * LD1;        // n * HID
  float* traces = Hid + (size_t)n * HID;         // 4 * 8 * 3
  float* gp     = traces + 96;                   // 6 (+2 pad)
  float* w1t0   = gp + 8;                        // 128 * LD0
  float* w2t0   = w1t0 + 128 * LD0;              // 128 * 128
  float* w1t1   = w2t0 + 128 * 128;              // 128 * LD1
  float* w2t1   = w1t1 + 128 * LD1;              // 128 * 128

  zero_f32<<<512, 256, 0, stream>>>(Phi0, n * LD0);
  zero_f32<<<512, 256, 0, stream>>>(Phi1, n * LD1);
  filter_prep<<<1, 32, 0, stream>>>(Wt0, Wc0, g0, Wt1, Wc1, g1, traces, gp);

  // weights: transpose + zero-pad for the WMMA GEMMs
  pad_transpose<<<(128 * LD0 + 255) / 256, 256, 0, stream>>>(w10, NF0, HID, LD0, w1t0);
  pad_transpose<<<(128 * 128 + 255) / 256, 256, 0, stream>>>(w20, HID, D, HID, w2t0);
  pad_transpose<<<(128 * LD1 + 255) / 256, 256, 0, stream>>>(w11, NF1, HID, LD1, w1t1);
  pad_transpose<<<(128 * 128 + 255) / 256, 256, 0, stream>>>(w21, HID, D, HID, w2t1);

  // degree feature: Phi0 col16, Phi1 col144, out col256
  row_sum<<<n, 256, 0, stream>>>(A, n, Phi0, LD0, 16, Phi1, LD1, 144, out, OLD, 2 * D);

  // motif channels (both layers in one pass each)
  motif_kernel<3><<<(I_tri + 255) / 256, 256, 0, stream>>>(
      A, n, tri, I_tri, traces + 0, gp, traces + 48, gp + 3, Phi0, LD0, 0, Phi1, LD1, 0);
  motif_kernel<4><<<(I_cyc + 255) / 256, 256, 0, stream>>>(
      A, n, cyc, I_cyc, traces + 24, gp, traces + 72, gp + 3, Phi0, LD0, 8, Phi1, LD1, 8);

  // ---- layer 0
  layernorm_rows<<<n, 32, 0, stream>>>(Phi0, LD0, NF0, lng0, lnb0);
  gemm_wmma_gelu<<<dim3(n / 16, HID / 16), 32, 0, stream>>>(Phi0, LD0, w1t0, LD0, b10,
                                                            Hid, HID);
  // Z0 -> out cols [128,256) and Phi1 cols [16,144)
  gemm_wmma_bias<<<dim3(n / 16, D / 16), 32, 0, stream>>>(Hid, HID, w2t0, HID, b20,
                                                          out, OLD, D, Phi1, LD1, 16);
  // ---- layer 1
  layernorm_rows<<<n, 32, 0, stream>>>(Phi1, LD1, NF1, lng1, lnb1);
  gemm_wmma_gelu<<<dim3(n / 16, HID / 16), 32, 0, stream>>>(Phi1, LD1, w1t1, LD1, b11,
                                                            Hid, HID);
  // Z1 -> out cols [0,128)
  gemm_wmma_bias<<<dim3(n / 16, D / 16), 32, 0, stream>>>(Hid, HID, w2t1, HID, b21,
                                                          out, OLD, 0, (float*)nullptr, 0, 0);
}